// Fastsum_87488483819775
// MI455X (gfx1250) — compile-verified
//
#include <hip/hip_runtime.h>
#include <math.h>

#define N_SRC   8192
#define M_TGT   8192
#define DIMS    64
#define P_SL    64
#define K_FT    1024
#define X_RANGE 0.3f

typedef float v2f __attribute__((ext_vector_type(2)));
typedef float v8f __attribute__((ext_vector_type(8)));

// ---- workspace layout (in floats) ----
#define WS_NORM  0                         // max row norm, stored as float bits (uint atomicMax)
#define WS_SF    1                         // sf = X_RANGE / norm_max
#define WS_KFT   16                        // K_FT entries
#define WS_APROJ 2048                      // [p][n] : P_SL x N_SRC (slice-major, unit stride in n)
#define WS_BPROJ (WS_APROJ + P_SL*N_SRC)   // [p][m] : P_SL x M_TGT
#define WS_CRE   (WS_BPROJ + P_SL*M_TGT)   // [p][k] : P_SL x K_FT
#define WS_CIM   (WS_CRE  + P_SL*K_FT)

// sin/cos of 2*pi*t : v_sin_f32 / v_cos_f32 take revolutions; reduce to [-0.5,0.5]
__device__ __forceinline__ void sincos2pi(float t, float& s, float& c) {
    float tr = t - rintf(t);
    s = __builtin_amdgcn_sinf(tr);
    c = __builtin_amdgcn_cosf(tr);
}

__device__ __forceinline__ v8f wmma_f32(v2f a, v2f b, v8f c) {
    // V_WMMA_F32_16X16X4_F32 : D = A(16x4) x B(4x16) + C(16x16), all f32
    return __builtin_amdgcn_wmma_f32_16x16x4_f32(false, a, false, b, (short)0, c,
                                                 false, false);
}

// ---------------- init: zero out + c accumulators + norm slot ----------------
__global__ void init_k(float* out, float* ws) {
    int i = blockIdx.x * blockDim.x + threadIdx.x;
    if (i < M_TGT) out[i] = 0.0f;
    if (i < 2 * P_SL * K_FT) ws[WS_CRE + i] = 0.0f;
    if (i == 0) ws[WS_NORM] = 0.0f;
}

// ---------------- max row norm over x and y ----------------
__global__ void norm_k(const float* __restrict__ x, const float* __restrict__ y,
                       float* ws) {
    int row = blockIdx.x * blockDim.x + threadIdx.x;       // 0 .. N+M-1
    const float* src = (row < N_SRC) ? (x + row * DIMS) : (y + (row - N_SRC) * DIMS);
    float s = 0.0f;
    #pragma unroll 8
    for (int j = 0; j < DIMS; ++j) { float v = src[j]; s += v * v; }
    float n = sqrtf(s);
    // positive floats order-preserving as uints
    atomicMax((unsigned int*)&ws[WS_NORM], __float_as_uint(n));
}

// ---------------- sf + Fourier coefficients of sliced Gaussian ----------------
__global__ void kft_k(const float* __restrict__ scale, float* ws) {
    int k = threadIdx.x;                                   // 0..1023
    float nmax = __uint_as_float(((unsigned int*)ws)[WS_NORM]);
    float sf = X_RANGE / nmax;
    if (k == 0) ws[WS_SF] = sf;
    float sr = scale[0] * sf;
    float s2 = sr * sr;
    float h  = (float)(k - K_FT / 2);
    float habs = fabsf(h);
    float v = 0.0f;
    if (habs > 0.0f) {
        const float TWO_PI2 = 19.739208802178716f;         // 2*pi^2
        float logv = 32.0f * logf(TWO_PI2 * s2)            // (d/2) log(2 pi^2 s2), d=64
                   + 63.0f * logf(habs)                    // (d-1) log|h|
                   - TWO_PI2 * s2 * h * h
                   - lgammaf(32.0f);                       // lgamma(d/2)
        v = expf(logv);
    }
    ws[WS_KFT + k] = v;
}

// ---------------- projections: A = X @ Xis^T via f32 WMMA, stored [p][n] -----
__global__ void proj_k(const float* __restrict__ x, const float* __restrict__ y,
                       const float* __restrict__ xis, float* ws) {
    int wave = (blockIdx.x * blockDim.x + threadIdx.x) >> 5;   // global wave id
    int lane = threadIdx.x & 31;
    int l    = lane & 15;
    int kadd = (lane >> 4) * 2;

    const float* src = x;
    float* dst = ws + WS_APROJ;
    int t = wave;                               // 4096 tiles: 2048 for x, 2048 for y
    if (t >= 2048) { src = y; dst = ws + WS_BPROJ; t -= 2048; }
    int m0 = (t >> 2) * 16;                     // point-row tile
    int n0 = (t & 3) * 16;                      // slice-col tile

    v8f c = {};
    #pragma unroll
    for (int k0 = 0; k0 < DIMS; k0 += 4) {
        v2f a, b;
        a.x = src[(m0 + l) * DIMS + k0 + kadd];
        a.y = src[(m0 + l) * DIMS + k0 + kadd + 1];
        b.x = xis[(n0 + l) * DIMS + k0 + kadd];     // B[k,n] = xis[n,k]
        b.y = xis[(n0 + l) * DIMS + k0 + kadd + 1];
        c = wmma_f32(a, b, c);
    }
    int mrow = m0 + ((lane >> 4) << 3);
    #pragma unroll
    for (int v = 0; v < 8; ++v)                 // transposed store: [p][n]
        dst[(n0 + l) * N_SRC + (mrow + v)] = c[v];
}

// ---------------- adjoint NDFT as complex GEMM 32x32x8192 (per slice) --------
// yhat[q,r] = sum_n U[q,n] * (w_n V[r,n]),  U = cis(2pi(32q-512)a_n), V = cis(2pi r a_n)
__global__ void adjoint_k(const float* __restrict__ xw, float* ws) {
    __shared__ float lre[K_FT];
    __shared__ float lim[K_FT];
    int p      = blockIdx.y;
    int chunk  = blockIdx.x;                    // 0..7, 1024 source points each
    int tid    = threadIdx.x;
    int waveId = tid >> 5;
    int lane   = tid & 31;
    int l      = lane & 15;
    int kadd   = (lane >> 4) * 2;
    float sf = ws[WS_SF];
    const float* aprj = ws + WS_APROJ + p * N_SRC;

    for (int i = tid; i < K_FT; i += 256) { lre[i] = 0.0f; lim[i] = 0.0f; }
    __syncthreads();

    v8f acc_re[2][2] = {};                      // [qtile][rtile]
    v8f acc_im[2][2] = {};

    int base = chunk * 1024 + waveId * 128;     // 32 K-steps of 4 points per wave
    for (int s = 0; s < 32; ++s) {
        int n1 = base + s * 4 + kadd;
        float a1 = sf * aprj[n1];
        float a2 = sf * aprj[n1 + 1];
        float w1 = xw[n1], w2 = xw[n1 + 1];

        v2f Ur[2], Ui[2], Uin[2];
        #pragma unroll
        for (int qt = 0; qt < 2; ++qt) {
            float f = (float)(32 * (qt * 16 + l) - 512);
            float s1, c1, s2, c2;
            sincos2pi(f * a1, s1, c1);
            sincos2pi(f * a2, s2, c2);
            Ur[qt].x = c1;  Ur[qt].y = c2;
            Ui[qt].x = s1;  Ui[qt].y = s2;
            Uin[qt].x = -s1; Uin[qt].y = -s2;
        }
        v2f Wr[2], Wi[2];
        #pragma unroll
        for (int rt = 0; rt < 2; ++rt) {
            float r = (float)(rt * 16 + l);
            float s1, c1, s2, c2;
            sincos2pi(r * a1, s1, c1);
            sincos2pi(r * a2, s2, c2);
            Wr[rt].x = w1 * c1; Wr[rt].y = w2 * c2;
            Wi[rt].x = w1 * s1; Wi[rt].y = w2 * s2;
        }
        #pragma unroll
        for (int qt = 0; qt < 2; ++qt)
            #pragma unroll
            for (int rt = 0; rt < 2; ++rt) {
                acc_re[qt][rt] = wmma_f32(Ur[qt],  Wr[rt], acc_re[qt][rt]);
                acc_re[qt][rt] = wmma_f32(Uin[qt], Wi[rt], acc_re[qt][rt]); // -Ui*Wi
                acc_im[qt][rt] = wmma_f32(Ur[qt],  Wi[rt], acc_im[qt][rt]);
                acc_im[qt][rt] = wmma_f32(Ui[qt],  Wr[rt], acc_im[qt][rt]);
            }
    }

    int half8 = (lane >> 4) * 8;
    #pragma unroll
    for (int qt = 0; qt < 2; ++qt)
        #pragma unroll
        for (int rt = 0; rt < 2; ++rt)
            #pragma unroll
            for (int v = 0; v < 8; ++v) {
                int q = qt * 16 + v + half8;
                int r = rt * 16 + l;
                atomicAdd(&lre[q * 32 + r], acc_re[qt][rt][v]);
                atomicAdd(&lim[q * 32 + r], acc_im[qt][rt][v]);
            }
    __syncthreads();
    for (int i = tid; i < K_FT; i += 256) {
        atomicAdd(&ws[WS_CRE + p * K_FT + i], lre[i]);
        atomicAdd(&ws[WS_CIM + p * K_FT + i], lim[i]);
    }
}

// ---------------- c[k] = kft[k] * yhat[k] ----------------
__global__ void scalec_k(float* ws) {
    int i = blockIdx.x * blockDim.x + threadIdx.x;   // over P_SL*K_FT
    int k = i & (K_FT - 1);
    float g = ws[WS_KFT + k];
    ws[WS_CRE + i] *= g;
    ws[WS_CIM + i] *= g;
}

// ---------------- forward NDFT: T = c^T @ conj(Ub) (WMMA), then r-contraction -
__global__ void forward_k(float* __restrict__ out, float* ws) {
    __shared__ float cre[K_FT];
    __shared__ float cim[K_FT];
    int p      = blockIdx.y;
    int mchunk = blockIdx.x;                    // 0..31, 256 targets each
    int tid    = threadIdx.x;
    int waveId = tid >> 5;
    int lane   = tid & 31;
    int l      = lane & 15;
    int kadd   = (lane >> 4) * 2;
    float sf = ws[WS_SF];
    const float* bprj = ws + WS_BPROJ + p * M_TGT;

    for (int i = tid; i < K_FT; i += 256) {
        cre[i] = ws[WS_CRE + p * K_FT + i];
        cim[i] = ws[WS_CIM + p * K_FT + i];
    }
    __syncthreads();

    for (int t = 0; t < 2; ++t) {               // 2 target tiles of 16 per wave
        int m0 = mchunk * 256 + waveId * 32 + t * 16;
        int m  = m0 + l;
        float b = sf * bprj[m];

        v8f Tre[2] = {}, Tim[2] = {};           // [rtile]
        #pragma unroll
        for (int k0 = 0; k0 < 32; k0 += 4) {    // K = q dimension
            int q1 = k0 + kadd;
            float f1 = (float)(32 * q1 - 512);
            float f2 = f1 + 32.0f;
            float sb1, cb1, sb2, cb2;
            sincos2pi(f1 * b, sb1, cb1);
            sincos2pi(f2 * b, sb2, cb2);
            v2f Bc = { cb1, cb2 };              // conj(U): re part
            v2f Bs = { sb1, sb2 };              // sin (sign handled in combos)
            #pragma unroll
            for (int rt = 0; rt < 2; ++rt) {
                int r = rt * 16 + l;
                v2f Are  = {  cre[q1 * 32 + r],  cre[(q1 + 1) * 32 + r] };
                v2f Aim  = {  cim[q1 * 32 + r],  cim[(q1 + 1) * 32 + r] };
                v2f Aren = { -Are.x, -Are.y };
                // T = c * conj(U): re = c_re*cb + c_im*sb ; im = c_im*cb - c_re*sb
                Tre[rt] = wmma_f32(Are,  Bc, Tre[rt]);
                Tre[rt] = wmma_f32(Aim,  Bs, Tre[rt]);
                Tim[rt] = wmma_f32(Aim,  Bc, Tim[rt]);
                Tim[rt] = wmma_f32(Aren, Bs, Tim[rt]);
            }
        }
        // s_m = Re( sum_r conj(V[r,m]) * T[r,m] ) = sum_r cos(2pi r b)*Tre + sin(2pi r b)*Tim
        float partial = 0.0f;
        int half8 = (lane >> 4) * 8;
        #pragma unroll
        for (int rt = 0; rt < 2; ++rt)
            #pragma unroll
            for (int v = 0; v < 8; ++v) {
                int r = rt * 16 + v + half8;
                float sr, cr;
                sincos2pi((float)r * b, sr, cr);
                partial += cr * Tre[rt][v] + sr * Tim[rt][v];
            }
        partial += __shfl_xor(partial, 16);     // lanes l and l+16 hold same target m
        if (lane < 16) atomicAdd(&out[m], partial * (1.0f / P_SL));
    }
}

extern "C" void kernel_launch(void* const* d_in, const int* in_sizes, int n_in,
                              void* d_out, int out_size, void* d_ws, size_t ws_size,
                              hipStream_t stream) {
    const float* x     = (const float*)d_in[0];
    const float* y     = (const float*)d_in[1];
    const float* xw    = (const float*)d_in[2];
    const float* scale = (const float*)d_in[3];
    const float* xis   = (const float*)d_in[4];
    float* out = (float*)d_out;
    float* ws  = (float*)d_ws;

    init_k   <<< (2 * P_SL * K_FT + 255) / 256, 256, 0, stream >>> (out, ws);
    norm_k   <<< (N_SRC + M_TGT) / 256,          256, 0, stream >>> (x, y, ws);
    kft_k    <<< 1, K_FT,                              0, stream >>> (scale, ws);
    proj_k   <<< 512, 256,                             0, stream >>> (x, y, xis, ws);
    adjoint_k<<< dim3(8, P_SL), 256,                   0, stream >>> (xw, ws);
    scalec_k <<< (P_SL * K_FT) / 256, 256,             0, stream >>> (ws);
    forward_k<<< dim3(32, P_SL), 256,                  0, stream >>> (out, ws);
}